// Render_12807592477417
// MI455X (gfx1250) — compile-verified
//
#include <hip/hip_runtime.h>
#include <float.h>

typedef __attribute__((ext_vector_type(2))) float v2f;
typedef __attribute__((ext_vector_type(4))) float v4f;
typedef __attribute__((ext_vector_type(8))) float v8f;

#define IMG 256
#define NTRI 256
#define TEX 512
#define TC_STRIDE 32     // floats per triangle record (padded for <<5 addressing)
#define NPIX (IMG * IMG)

#if __has_builtin(__builtin_amdgcn_rcpf)
#define FAST_RCP(x) __builtin_amdgcn_rcpf(x)
#else
#define FAST_RCP(x) (1.f / (x))
#endif

// Per-triangle LDS record (offsets in floats):
//  0..3  : edge AB  [cx, cy, c0, 0]
//  4..7  : edge CB  [cx, cy, c0, 0]
//  8..11 : edge CA  [cx, cy, c0, 0]
// 12..15 : z-plane  [cx, cy, c0, 0]   (z*w numerator)
// 16..20 : w, mnx, mxx, mny, mxy
// 21     : pad
// 22..27 : u0x,u0y,u1x,u1y,u2x,u2y    (uv*2-1)
// 28..31 : pad

__global__ __launch_bounds__(256) void raster_wmma_kernel(
    const float* __restrict__ tris,   // [256,3,3]
    const float* __restrict__ uvs,    // [256,3,2]
    const float* __restrict__ uvmap,  // [3,512,512]
    float* __restrict__ out)          // [4,256,256]
{
    __shared__ float tc[NTRI * TC_STRIDE];
    __shared__ float red[256];

    const int tid = threadIdx.x;

    // warm L2 with the texture (global_prefetch_b8 path)
    __builtin_prefetch(&uvmap[(size_t)tid * 3072u], 0, 3);

    // ---- stage + precompute per-triangle constants (1 thread : 1 triangle) ----
    {
        const int t = tid;
        const float ax = tris[t * 9 + 0], ay = tris[t * 9 + 1], az = tris[t * 9 + 2];
        const float bx = tris[t * 9 + 3], by = tris[t * 9 + 4], bz = tris[t * 9 + 5];
        const float cx = tris[t * 9 + 6], cy = tris[t * 9 + 7], cz = tris[t * 9 + 8];
        float* p = &tc[t * TC_STRIDE];
        // pAB = area2d(B, pt, A)
        const float cxAB = ay - by, cyAB = bx - ax;
        const float c0AB = by * (ax - bx) - bx * (ay - by);
        // pCB = area2d(C, pt, B)
        const float cxCB = by - cy, cyCB = cx - bx;
        const float c0CB = cy * (bx - cx) - cx * (by - cy);
        // pCA = area2d(A, pt, C)
        const float cxCA = cy - ay, cyCA = ax - cx;
        const float c0CA = ay * (cx - ax) - ax * (cy - ay);
        p[0] = cxAB; p[1] = cyAB; p[2]  = c0AB; p[3]  = 0.f;
        p[4] = cxCB; p[5] = cyCB; p[6]  = c0CB; p[7]  = 0.f;
        p[8] = cxCA; p[9] = cyCA; p[10] = c0CA; p[11] = 0.f;
        // z numerator plane: z*w = pCB*z0 + pCA*z1 + pAB*z2
        p[12] = cxCB * az + cxCA * bz + cxAB * cz;
        p[13] = cyCB * az + cyCA * bz + cyAB * cz;
        p[14] = c0CB * az + c0CA * bz + c0AB * cz;
        p[15] = 0.f;
        p[16] = (bx - ax) * (cy - ay) - (by - ay) * (cx - ax);   // w
        p[17] = fminf(ax, fminf(bx, cx));  p[18] = fmaxf(ax, fmaxf(bx, cx));
        p[19] = fminf(ay, fminf(by, cy));  p[20] = fmaxf(ay, fmaxf(by, cy));
        p[21] = 0.f;
        p[22] = uvs[t * 6 + 0] * 2.f - 1.f; p[23] = uvs[t * 6 + 1] * 2.f - 1.f;
        p[24] = uvs[t * 6 + 2] * 2.f - 1.f; p[25] = uvs[t * 6 + 3] * 2.f - 1.f;
        p[26] = uvs[t * 6 + 4] * 2.f - 1.f; p[27] = uvs[t * 6 + 5] * 2.f - 1.f;
        p[28] = 0.f; p[29] = 0.f; p[30] = 0.f; p[31] = 0.f;
        red[tid] = fminf(az, fminf(bz, cz));
    }
    __syncthreads();
    // block reduce: zInit = min over all vertex z  (== reference zbuf init)
    for (int s = 128; s > 0; s >>= 1) {
        if (tid < s) red[tid] = fminf(red[tid], red[tid + s]);
        __syncthreads();
    }
    const float zInit = red[0];

    // ---- wave / pixel mapping: each wave32 owns 16 consecutive pixels ----
    const int wave = tid >> 5;
    const int lane = tid & 31;
    const int half = lane >> 4;   // D rows 0..7 (half 0) / 8..15 (half 1)
    const int tl   = lane & 15;   // triangle slot within a batch of 16
    const int pixBase = blockIdx.x * 128 + wave * 16;

    const int row = pixBase >> 8;
    const int colBase = pixBase & 255;
    const float y = -1.f + (2.f / 255.f) * (float)(255 - row);

    // A matrix (16x4): lanes 0..15 -> {x,y} (K=0,1) of pixel M=lane;
    //                  lanes 16..31 -> {1,0} (K=2,3)
    v2f amat;
    amat.x = (half == 0) ? (-1.f + (2.f / 255.f) * (float)(colBase + tl)) : 1.f;
    amat.y = (half == 0) ? y : 0.f;

    float bestZ[8];
    int bestIdx[8];
#pragma unroll
    for (int v = 0; v < 8; ++v) { bestZ[v] = -FLT_MAX; bestIdx[v] = -1; }

    // ---- 16 batches x 16 triangles; 4 WMMAs: 3 edges + z-plane ----
#pragma unroll 2
    for (int batch = 0; batch < 16; ++batch) {
        const int t = batch * 16 + tl;
        const float* p = &tc[t * TC_STRIDE];
        const int bo = half * 2;   // half 0 -> {cx,cy}; half 1 -> {c0,0}

        const v2f bAB = *(const v2f*)(p + 0  + bo);
        const v2f bCB = *(const v2f*)(p + 4  + bo);
        const v2f bCA = *(const v2f*)(p + 8  + bo);
        const v2f bZ  = *(const v2f*)(p + 12 + bo);

        v8f zero = {};
        const v8f eAB = __builtin_amdgcn_wmma_f32_16x16x4_f32(false, amat, false, bAB,
                                                              (short)0, zero, false, false);
        const v8f eCB = __builtin_amdgcn_wmma_f32_16x16x4_f32(false, amat, false, bCB,
                                                              (short)0, zero, false, false);
        const v8f eCA = __builtin_amdgcn_wmma_f32_16x16x4_f32(false, amat, false, bCA,
                                                              (short)0, zero, false, false);
        const v8f zn  = __builtin_amdgcn_wmma_f32_16x16x4_f32(false, amat, false, bZ,
                                                              (short)0, zero, false, false);

        const v4f s0  = *(const v4f*)(p + 16);   // w, mnx, mxx, mny
        const float mxy = p[20];
        const float w   = s0.x;
        const bool  valid = (w >= 1e-9f);
        const float sw   = (w != 0.f) ? w : 1.f;  // safe_w
        const float invW = FAST_RCP(sw);          // v_rcp_f32, no IEEE div chain
        const bool  bby  = (y >= s0.w) && (y <= mxy);

#pragma unroll
        for (int v = 0; v < 8; ++v) {
            const float x = -1.f + (2.f / 255.f) * (float)(colBase + half * 8 + v);
            const bool bb = bby && (x >= s0.y) && (x <= s0.z);
            const bool inside = (eAB[v] > 0.f) && (eCB[v] > 0.f) && (eCA[v] > 0.f);
            const float z = zn[v] * invW;
            // sequential z>=zbuf scan == lexicographic (z, idx) argmax,
            // ties won by later index; idx increases within a lane.
            const bool cand = valid && bb && inside && (z >= zInit) && (z >= bestZ[v]);
            bestZ[v]   = cand ? z : bestZ[v];
            bestIdx[v] = cand ? t : bestIdx[v];
        }
    }

    // ---- butterfly (z, idx) argmax over the 16 triangle-lanes of each half ----
#pragma unroll
    for (int m = 1; m <= 8; m <<= 1) {
#pragma unroll
        for (int v = 0; v < 8; ++v) {
            const float oz = __shfl_xor(bestZ[v], m, 32);
            const int   oi = __shfl_xor(bestIdx[v], m, 32);
            const bool take = (oi >= 0) &&
                              ((bestIdx[v] < 0) || (oz > bestZ[v]) ||
                               (oz == bestZ[v] && oi > bestIdx[v]));
            bestZ[v]   = take ? oz : bestZ[v];
            bestIdx[v] = take ? oi : bestIdx[v];
        }
    }

    // ---- shade + store: lanes 0-7 -> pixels 0-7, lanes 16-23 -> pixels 8-15 ----
    if (tl < 8) {
        const int v = tl;
        const int col = colBase + half * 8 + v;
        const int pix = pixBase + half * 8 + v;
        float r = 0.f, g = 0.f, b = 0.f, a = 0.f;
        if (bestIdx[v] >= 0) {
            const float* q = &tc[bestIdx[v] * TC_STRIDE];
            const float x = -1.f + (2.f / 255.f) * (float)col;
            const float w = q[16];
            const float sw = (w != 0.f) ? w : 1.f;
            // exact IEEE division here to mirror the reference barycentrics
            const float w1 = (q[4] * x + q[5] * y + q[6])  / sw;   // pCB / safe_w
            const float w2 = (q[8] * x + q[9] * y + q[10]) / sw;   // pCA / safe_w
            const float w3 = 1.f - w1 - w2;
            const float gu = w1 * q[22] + w2 * q[24] + w3 * q[26];
            const float gv = w1 * q[23] + w2 * q[25] + w3 * q[27];
            // grid_sample bilinear, align_corners=False, zero padding
            const float xs = ((gu + 1.f) * (float)TEX - 1.f) * 0.5f;
            const float ys = ((gv + 1.f) * (float)TEX - 1.f) * 0.5f;
            const float x0 = floorf(xs), y0 = floorf(ys);
            const float wx = xs - x0, wy = ys - y0;

            float cr[4], cg[4], cb[4];
            const float cxs[4] = { x0, x0 + 1.f, x0, x0 + 1.f };
            const float cys[4] = { y0, y0, y0 + 1.f, y0 + 1.f };
#pragma unroll
            for (int k = 0; k < 4; ++k) {
                const float xf = cxs[k], yf = cys[k];
                const bool inb = (xf >= 0.f) && (xf <= (float)(TEX - 1)) &&
                                 (yf >= 0.f) && (yf <= (float)(TEX - 1));
                const int xi = (int)fminf(fmaxf(xf, 0.f), (float)(TEX - 1));
                const int yi = (int)fminf(fmaxf(yf, 0.f), (float)(TEX - 1));
                const float s = inb ? 1.f : 0.f;
                const int o = yi * TEX + xi;
                cr[k] = uvmap[o] * s;
                cg[k] = uvmap[TEX * TEX + o] * s;
                cb[k] = uvmap[2 * TEX * TEX + o] * s;
            }
            const float w00 = (1.f - wx) * (1.f - wy), w10 = wx * (1.f - wy);
            const float w01 = (1.f - wx) * wy,         w11 = wx * wy;
            r = cr[0] * w00 + cr[1] * w10 + cr[2] * w01 + cr[3] * w11;
            g = cg[0] * w00 + cg[1] * w10 + cg[2] * w01 + cg[3] * w11;
            b = cb[0] * w00 + cb[1] * w10 + cb[2] * w01 + cb[3] * w11;
            a = 1.f;
        }
        out[0 * NPIX + pix] = r;
        out[1 * NPIX + pix] = g;
        out[2 * NPIX + pix] = b;
        out[3 * NPIX + pix] = a;
    }
}

extern "C" void kernel_launch(void* const* d_in, const int* in_sizes, int n_in,
                              void* d_out, int out_size, void* d_ws, size_t ws_size,
                              hipStream_t stream) {
    const float* tris  = (const float*)d_in[0];   // [256,3,3]
    const float* uvs   = (const float*)d_in[1];   // [256,3,2]
    const float* uvmap = (const float*)d_in[2];   // [3,512,512]
    float* out = (float*)d_out;                   // [4,256,256]
    (void)in_sizes; (void)n_in; (void)out_size; (void)d_ws; (void)ws_size;

    dim3 grid(NPIX / 128);   // 512 blocks, 128 pixels each
    dim3 block(256);         // 8 wave32, 16 pixels per wave
    hipLaunchKernelGGL(raster_wmma_kernel, grid, block, 0, stream,
                       tris, uvs, uvmap, out);
}